// Group_28398323761380
// MI455X (gfx1250) — compile-verified
//
#include <hip/hip_runtime.h>

#define Bb 8
#define Nn 8192
#define Ss 2048
#define Dd 64
#define Kk 16
#define COUT (3 + Dd)            // 67 output channels
#define CHUNK 512                // points per LDS distance chunk
#define DSTRIDE (CHUNK + 4)      // pad rows: 516 % 64 = 4 -> conflict-free lanes
#define NTHREADS 128
#define NWAVES (NTHREADS / 32)

typedef __attribute__((ext_vector_type(2))) float v2f;
typedef __attribute__((ext_vector_type(4))) float v4f;
typedef __attribute__((ext_vector_type(8))) float v8f;

__global__ __launch_bounds__(NTHREADS) void knn_group_kernel(
    const float* __restrict__ points,      // [B,3,N]
    const float* __restrict__ new_points,  // [B,3,S]
    const float* __restrict__ features,    // [B,D,N]
    float* __restrict__ out)               // [B,67,K,S]
{
  __shared__ __align__(16) float s_dist[16 * DSTRIDE];  // 16 queries x padded chunk
  __shared__ __align__(16) float s_cand_d[16 * 128];    // merge buffers
  __shared__ __align__(16) int   s_cand_i[16 * 128];
  __shared__ int   s_topk[16 * Kk];

  const int b    = blockIdx.y;
  const int qt   = blockIdx.x;             // tile of 16 queries
  const int tid  = threadIdx.x;
  const int lane = tid & 31;
  // force wave index into an SGPR so the tile loop is scalar-branched
  const int wave = __builtin_amdgcn_readfirstlane(tid >> 5);
  const int half = lane >> 4;              // 0: lanes 0-15, 1: lanes 16-31
  const int l16  = lane & 15;

  const float* pB = points     + (size_t)b * 3  * Nn;
  const float* qB = new_points + (size_t)b * 3  * Ss;
  const float* fB = features   + (size_t)b * Dd * Nn;

  // ---- B matrix (queries, 4x16) and C matrix (query norms): block-invariant
  const int s_idx = qt * 16 + l16;
  const float qx = qB[0 * Ss + s_idx];
  const float qy = qB[1 * Ss + s_idx];
  const float qz = qB[2 * Ss + s_idx];
  const float qn = qx * qx + qy * qy + qz * qz;
  v2f bm;
  bm.x = half ? qz   : qx;                 // B rows: K0=qx K1=qy K2=qz K3=1
  bm.y = half ? 1.0f : qy;
  v8f cm;
  #pragma unroll
  for (int v = 0; v < 8; ++v) cm[v] = qn;  // C[m][n] = ||q_n||^2

  // per-thread running top-K (8 scan threads per query)
  const int q_of = tid >> 3;               // query 0..15
  const int part = tid & 7;                // 1/8th of chunk
  float bestd[Kk];
  int   besti[Kk];
  #pragma unroll
  for (int j = 0; j < Kk; ++j) { bestd[j] = 3.4e38f; besti[j] = 0; }

  for (int base = 0; base < Nn; base += CHUNK) {
    // ---- WMMA distance tiles: D = A(16x4) x B(4x16) + C ----
    #pragma unroll 2
    for (int t = wave; t < CHUNK / 16; t += NWAVES) {
      const int pidx = base + t * 16 + l16;
      const float px = pB[0 * Nn + pidx];
      const float py = pB[1 * Nn + pidx];
      const float pz = pB[2 * Nn + pidx];
      const float pn = px * px + py * py + pz * pz;
      v2f am;                               // A row m: [-2x,-2y,-2z,||p||^2]
      am.x = half ? (-2.0f * pz) : (-2.0f * px);
      am.y = half ? pn           : (-2.0f * py);
      v8f dm = __builtin_amdgcn_wmma_f32_16x16x4_f32(
          false, am, false, bm, (short)0, cm, false, false);
      // lane holds column n=l16; rows m = half*8 + v  (contiguous -> b128 x2)
      float* row = &s_dist[l16 * DSTRIDE + t * 16 + half * 8];
      #pragma unroll
      for (int v = 0; v < 8; ++v) row[v] = dm[v];
    }
    if (base + CHUNK < Nn) {               // prefetch next chunk's coords
      const int nidx = base + CHUNK + tid * 4;
      __builtin_prefetch(&pB[0 * Nn + nidx], 0, 0);
      __builtin_prefetch(&pB[1 * Nn + nidx], 0, 0);
      __builtin_prefetch(&pB[2 * Nn + nidx], 0, 0);
    }
    __syncthreads();

    // ---- top-K scan: b128 LDS reads + branch-free register insertion ----
    const v4f* drow4 = (const v4f*)&s_dist[q_of * DSTRIDE + part * (CHUNK / 8)];
    #pragma unroll 2
    for (int j4 = 0; j4 < (CHUNK / 8) / 4; ++j4) {
      const v4f dv = drow4[j4];
      #pragma unroll
      for (int e = 0; e < 4; ++e) {
        const float d = dv[e];
        if (d < bestd[Kk - 1]) {
          float cd = d;
          int   ci = base + part * (CHUNK / 8) + j4 * 4 + e;
          #pragma unroll
          for (int u = 0; u < Kk; ++u) {
            const bool lt = cd < bestd[u];
            const float tf = bestd[u]; const int ti = besti[u];
            if (lt) { bestd[u] = cd; besti[u] = ci; cd = tf; ci = ti; }
          }
        }
      }
    }
    __syncthreads();
  }

  // ---- merge 8 partial lists -> final top-16 per query ----
  #pragma unroll
  for (int j = 0; j < Kk; ++j) {
    s_cand_d[q_of * 128 + part * Kk + j] = bestd[j];
    s_cand_i[q_of * 128 + part * Kk + j] = besti[j];
  }
  __syncthreads();
  if (tid < 16) {
    float fd[Kk]; int fi[Kk];
    #pragma unroll
    for (int j = 0; j < Kk; ++j) { fd[j] = 3.4e38f; fi[j] = 0; }
    for (int j = 0; j < 128; ++j) {
      const float d = s_cand_d[tid * 128 + j];
      if (d < fd[Kk - 1]) {
        float cd = d; int ci = s_cand_i[tid * 128 + j];
        #pragma unroll
        for (int u = 0; u < Kk; ++u) {
          const bool lt = cd < fd[u];
          const float tf = fd[u]; const int ti = fi[u];
          if (lt) { fd[u] = cd; fi[u] = ci; cd = tf; ci = ti; }
        }
      }
    }
    #pragma unroll
    for (int j = 0; j < Kk; ++j) s_topk[tid * Kk + j] = fi[j];
  }
  __syncthreads();

  // ---- gather & write out[b][c][k][s]; q fastest -> coalesced along S ----
  for (int i = tid; i < COUT * Kk * 16; i += NTHREADS) {
    const int q = i & 15;
    const int k = (i >> 4) & 15;
    const int c = i >> 8;
    const int s = qt * 16 + q;
    const int n = s_topk[q * Kk + k];
    float val;
    if (c < 3) val = pB[c * Nn + n] - qB[c * Ss + s];   // relative coords
    else       val = fB[(c - 3) * Nn + n];              // grouped features
    out[(((size_t)b * COUT + c) * Kk + k) * Ss + s] = val;
  }
}

extern "C" void kernel_launch(void* const* d_in, const int* in_sizes, int n_in,
                              void* d_out, int out_size, void* d_ws, size_t ws_size,
                              hipStream_t stream) {
  const float* points     = (const float*)d_in[0];
  const float* new_points = (const float*)d_in[1];
  const float* features   = (const float*)d_in[2];
  float* out = (float*)d_out;
  dim3 grid(Ss / 16, Bb);
  knn_group_kernel<<<grid, dim3(NTHREADS), 0, stream>>>(points, new_points, features, out);
}